// energyfunc_loss_46918222741915
// MI455X (gfx1250) — compile-verified
//
#include <hip/hip_runtime.h>
#include <math.h>

// ---------------------------------------------------------------------------
// Problem constants
// ---------------------------------------------------------------------------
#define N_CHAINS   2
#define CONF_LEN   16388
#define NSTEPS     8192            // steps per chain
#define NPTS       8193            // positions per chain (leading zero + cumsum)
#define NTILES     513             // ceil(8193/16)
#define NPAD       (NTILES * 16)   // 8208 padded points
#define ROWPAIRS   257             // rows 0..256 paired with 512..256
#define SUBSPLIT   8               // column striping per row-pair
#define WAVES_PER_CHAIN (ROWPAIRS * SUBSPLIT)        // 2056
#define TOTAL_WAVES     (N_CHAINS * WAVES_PER_CHAIN) // 4112
#define BLOCKS_P2       (TOTAL_WAVES / 8)            // 514 blocks of 256 thr
// exp(-d2/0.25) = exp2(-4*log2(e) * d2)
#define NEG4LOG2E  (-5.770780163555852f)
#define SIGMA_F    10.0f

typedef __attribute__((ext_vector_type(2))) float v2f;
typedef __attribute__((ext_vector_type(8))) float v8f;

// ---------------------------------------------------------------------------
// Phase 1: build positions. One 1024-thread block per chain.
// Writes two padded buffers:
//   P [j] = ( x,   y,   z,  |p|^2)   -> A fragments + sq_j scalar
//   PB[j] = (-2x, -2y, -2z,  1   )   -> B fragments, zero in-loop ALU
// ---------------------------------------------------------------------------
__global__ __launch_bounds__(1024)
void build_positions(const float* __restrict__ conf,
                     float4* __restrict__ pos, float4* __restrict__ posB)
{
    const int chain = blockIdx.x;
    const float* c = conf + chain * CONF_LEN + 4;
    float4* P  = pos  + chain * NPAD;
    float4* PB = posB + chain * NPAD;
    const int t = threadIdx.x;

    __shared__ float lx[1024], ly[1024], lz[1024];

    float px[8], py[8], pz[8];
    float sx = 0.f, sy = 0.f, sz = 0.f;
    #pragma unroll
    for (int s = 0; s < 8; ++s) {
        const int k  = t * 8 + s;
        const float r1 = c[2 * k]     + 0.5f;
        const float r2 = c[2 * k + 1] + 0.5f;
        const float ct = 1.0f - 2.0f * r2;                   // cos(theta)
        const float st = sqrtf(fmaxf(1.0f - ct * ct, 0.0f)); // sin(theta)>=0
        float sp, cp;
        __sincosf(6.2831853071795864f * r1, &sp, &cp);
        sx += st * cp; px[s] = sx;
        sy += st * sp; py[s] = sy;
        sz += ct;      pz[s] = sz;
    }

    // inclusive block scan of thread totals
    lx[t] = sx; ly[t] = sy; lz[t] = sz;
    __syncthreads();
    for (int off = 1; off < 1024; off <<= 1) {
        float ax = 0.f, ay = 0.f, az = 0.f;
        if (t >= off) { ax = lx[t - off]; ay = ly[t - off]; az = lz[t - off]; }
        __syncthreads();
        lx[t] += ax; ly[t] += ay; lz[t] += az;
        __syncthreads();
    }
    const float ex = lx[t] - sx, ey = ly[t] - sy, ez = lz[t] - sz; // exclusive

    if (t == 0) {
        P[0]  = make_float4(0.f, 0.f, 0.f, 0.f);
        PB[0] = make_float4(0.f, 0.f, 0.f, 1.f);
    }
    #pragma unroll
    for (int s = 0; s < 8; ++s) {
        const float x = ex + px[s], y = ey + py[s], z = ez + pz[s];
        const int idx = 1 + t * 8 + s;
        P[idx]  = make_float4(x, y, z, x * x + y * y + z * z);
        PB[idx] = make_float4(-2.f * x, -2.f * y, -2.f * z, 1.f);
    }
    if (t < NPAD - NPTS) {                                 // 15 pad entries
        P[NPTS + t]  = make_float4(0.f, 0.f, 0.f, 0.f);
        PB[NPTS + t] = make_float4(0.f, 0.f, 0.f, 1.f);
    }
}

// ---------------------------------------------------------------------------
// One 16x16 tile: D = A*B + C with C = sq_j broadcast  ->  D[m][n] = d2(i,j).
// Returns per-lane (column) partial sum of exp(-4*max(d2,0)) over valid rows.
// Clamp folded as min(exp2(k*d2), 1): exp2 is monotone and its result is
// canonical, so this is a single v_min_num_f32 per element.
// ---------------------------------------------------------------------------
__device__ __forceinline__ float tile_partial(
    const float* __restrict__ Pf, const float* __restrict__ PBf,
    v2f a, int i0, int half, int j)
{
    const float2 bv = *(const float2*)(PBf + (size_t)j * 4 + half * 2);
    v2f b; b[0] = bv.x; b[1] = bv.y;
    const float sqj = Pf[(size_t)j * 4 + 3];

    v8f cin;
    #pragma unroll
    for (int q = 0; q < 8; ++q) cin[q] = sqj;

    v8f d = __builtin_amdgcn_wmma_f32_16x16x4_f32(
        false, a, false, b, (short)0, cin, false, false);

    float p = 0.0f;
    #pragma unroll
    for (int q = 0; q < 8; ++q) {
        float e = __builtin_amdgcn_exp2f(d[q] * NEG4LOG2E);
        e = fminf(e, 1.0f);                     // == exp(-4*max(d2,0))
        const int i = i0 + q + half * 8;        // row mask: tj-loop invariant
        p += (i < NPTS) ? e : 0.0f;
    }
    return p;
}

// ---------------------------------------------------------------------------
// Phase 2: triangular all-pairs energy via V_WMMA_F32_16X16X4_F32.
// Wave-uniform control (readfirstlane) -> scalar loop, EXEC all-1s at WMMA.
// Diagonal tile (weight 1) and ragged tj==512 tile peeled; hot loop has
// constant weight 2 and no masks.
// ---------------------------------------------------------------------------
__global__ __launch_bounds__(256)
void pair_energy(const float4* __restrict__ pos,
                 const float4* __restrict__ posB,
                 float* __restrict__ partials)
{
    const int wave = __builtin_amdgcn_readfirstlane(
        (int)((blockIdx.x * blockDim.x + threadIdx.x) >> 5));
    const int lane = threadIdx.x & 31;
    if (wave >= TOTAL_WAVES) return;

    const int chain = wave / WAVES_PER_CHAIN;
    const int rem   = wave % WAVES_PER_CHAIN;
    const int r     = rem / SUBSPLIT;
    const int sub   = rem % SUBSPLIT;

    const float* Pf  = (const float*)(pos  + chain * NPAD);
    const float* PBf = (const float*)(posB + chain * NPAD);
    const int half = lane >> 4;        // 0: k=0,1   1: k=2,3
    const int lm   = lane & 15;        // row/col within tile

    float acc = 0.0f;

    #pragma unroll
    for (int pass = 0; pass < 2; ++pass) {
        const int ti = (pass == 0) ? r : (512 - r);
        if (pass == 1 && ti == r) break;          // r==256: single row

        const int i0 = ti * 16;
        // A fragment: two consecutive comps of (x,y,z,sq) of point i0+lm
        const float2 av = *(const float2*)(Pf + (size_t)(i0 + lm) * 4 + half * 2);
        v2f a; a[0] = av.x; a[1] = av.y;

        // --- diagonal tile, weight 1 (owned by sub==0) ---
        if (sub == 0) {
            const int j = i0 + lm;
            const float p = tile_partial(Pf, PBf, a, i0, half, j);
            acc += ((j < NPTS) ? 1.0f : 0.0f) * p;
        }

        const int start = ti + (sub == 0 ? SUBSPLIT : sub);

        // --- hot loop: strictly off-diagonal, fully valid columns ---
        #pragma unroll 2
        for (int tj = start; tj < NTILES - 1; tj += SUBSPLIT) {
            const float p = tile_partial(Pf, PBf, a, i0, half, tj * 16 + lm);
            acc = fmaf(2.0f, p, acc);
        }

        // --- ragged last tile (tj == 512), off-diagonal here ---
        if (start <= NTILES - 1 && ((NTILES - 1 - start) % SUBSPLIT) == 0) {
            const int j = (NTILES - 1) * 16 + lm;
            const float p = tile_partial(Pf, PBf, a, i0, half, j);
            acc += 2.0f * ((j < NPTS) ? 1.0f : 0.0f) * p;
        }
    }

    // wave32 reduction
    #pragma unroll
    for (int o = 16; o > 0; o >>= 1) acc += __shfl_xor(acc, o, 32);
    if (lane == 0) partials[wave] = acc;
}

// ---------------------------------------------------------------------------
// Phase 3: deterministic fixed-order reduction of wave partials.
// ---------------------------------------------------------------------------
__global__ __launch_bounds__(256)
void final_reduce(const float* __restrict__ partials, int n,
                  float* __restrict__ out)
{
    __shared__ float s[256];
    const int t = threadIdx.x;
    float v = 0.0f;
    for (int i = t; i < n; i += 256) v += partials[i];
    s[t] = v;
    __syncthreads();
    for (int off = 128; off > 0; off >>= 1) {
        if (t < off) s[t] += s[t + off];
        __syncthreads();
    }
    if (t == 0) out[0] = SIGMA_F * s[0];
}

// ---------------------------------------------------------------------------
extern "C" void kernel_launch(void* const* d_in, const int* in_sizes, int n_in,
                              void* d_out, int out_size, void* d_ws, size_t ws_size,
                              hipStream_t stream)
{
    const float* conf = (const float*)d_in[0];
    float* out = (float*)d_out;

    // workspace layout
    const size_t posBytes = (size_t)N_CHAINS * NPAD * sizeof(float4);
    float4* pos      = (float4*)d_ws;
    float4* posB     = (float4*)((char*)d_ws + posBytes);
    float*  partials = (float*)((char*)d_ws + 2 * posBytes);

    build_positions<<<N_CHAINS, 1024, 0, stream>>>(conf, pos, posB);
    pair_energy<<<BLOCKS_P2, 256, 0, stream>>>(pos, posB, partials);
    final_reduce<<<1, 256, 0, stream>>>(partials, TOTAL_WAVES, out);
}